// AdaptiveChannelMixing_80109730005015
// MI455X (gfx1250) — compile-verified
//
#include <hip/hip_runtime.h>
#include <hip/hip_bf16.h>

// AdaptiveChannelMixing on MI455X (gfx1250), fp32 end-to-end.
// Stage 1: w[2400,16384] = qc[2400,256] @ W_gen[256,16384] + b  (v_wmma_f32_16x16x4_f32)
// Stage 2: per (b,n): mixed = xg @ w_g^T, LayerNorm(256), ReLU   (v_wmma_f32_16x16x4_f32)
// Staging copies use GLOBAL_LOAD_ASYNC_TO_LDS (ASYNCcnt) when the toolchain
// declares the builtin; otherwise fall back to manual coalesced staging.

typedef float v2f __attribute__((ext_vector_type(2)));
typedef float v8f __attribute__((ext_vector_type(8)));
typedef int   v4i __attribute__((vector_size(16)));   // matches builtin param type

#define HIDDEN   256
#define OUTDIM   16384   // 4 * 64 * 64
#define PNUM     32
#define MTOT     2400    // B*N = 8*300
#define LDSPAD   260     // 256 + 4 floats: bank-conflict-free column walks

#if defined(__has_builtin)
#if __has_builtin(__builtin_amdgcn_global_load_async_to_lds_b128)
#define HAVE_ASYNC_LDS 1
#endif
#endif
#ifndef HAVE_ASYNC_LDS
#define HAVE_ASYNC_LDS 0
#endif

#define GLOBAL_AS __attribute__((address_space(1)))
#define LDS_AS    __attribute__((address_space(3)))

// Generic->global: flat and global addresses are value-identical.
__device__ __forceinline__ GLOBAL_AS v4i* to_global_v4(const void* p) {
    return (GLOBAL_AS v4i*)(unsigned long long)p;
}
// Generic->LDS: generic LDS addr = {aperture, offset32}; low 32 bits are the
// wave-relative LDS byte offset (ISA 10.2 aperture rules).
__device__ __forceinline__ LDS_AS v4i* to_lds_v4(void* p) {
    return (LDS_AS v4i*)(unsigned int)(unsigned long long)p;
}

__device__ __forceinline__ void async_wait_all() {
    asm volatile("s_wait_asynccnt 0x0" ::: "memory");
}

__device__ __forceinline__ v8f wmma_f32_4(v2f a, v2f b, v8f c) {
    // D = A(16x4 f32) x B(4x16 f32) + C(16x16 f32)
    return __builtin_amdgcn_wmma_f32_16x16x4_f32(
        /*neg_a=*/false, a, /*neg_b=*/false, b,
        /*c_mod=*/(short)0, c, /*reuse_a=*/false, /*reuse_b=*/false);
}

// ---------------------------------------------------------------------------
// Stage 1: dynamic weight generation GEMM.
// grid = (OUTDIM/64, rows/16), block = 32 (one wave). Each wave: 16x64 tile.
// ---------------------------------------------------------------------------
__global__ __launch_bounds__(32) void acm_wgen_kernel(
    const float* __restrict__ qc,     // [2400, 256]
    const float* __restrict__ Wg,     // [256, 16384]
    const float* __restrict__ bg,     // [16384]
    float* __restrict__ w_out,        // [chunk_rows, 16384] (chunk-relative)
    int m0base)
{
    __shared__ float la[16 * LDSPAD];

    const int lane  = threadIdx.x;          // 0..31
    const int mrel0 = blockIdx.y * 16;      // row tile within chunk
    const int mabs0 = m0base + mrel0;       // absolute (b,n) row
    const int n0    = blockIdx.x * 64;      // output-column tile

    // Stage QC tile [16 x 256] into LDS.
#if HAVE_ASYNC_LDS
    // 16 KB = 1024 x 16B chunks; rows are 256 floats (64 chunks) -> chunks
    // never straddle the LDSPAD row padding.
    for (int ci = lane; ci < 1024; ci += 32) {
        const int f = ci * 4;                 // float index in compact tile
        const int r = f >> 8, c = f & 255;
        __builtin_amdgcn_global_load_async_to_lds_b128(
            to_global_v4(qc + (size_t)mabs0 * HIDDEN + f),
            to_lds_v4(&la[r * LDSPAD + c]), 0, 0);
    }
    async_wait_all();
#else
    #pragma unroll 4
    for (int i = lane; i < 16 * 256; i += 32) {
        const int r = i >> 8, c = i & 255;
        la[r * LDSPAD + c] = qc[(size_t)(mabs0 + r) * HIDDEN + c];
    }
#endif
    __syncthreads();

    const int row  = lane & 15;   // A: M index / B,D: N index
    const int half = lane >> 4;   // K / M-half selector

    v8f acc0{}, acc1{}, acc2{}, acc3{};

    #pragma unroll 4
    for (int k0 = 0; k0 < HIDDEN; k0 += 4) {
        // A fragment: A[m][k0 + 2*half], A[m][k0 + 2*half + 1] (8B LDS load)
        const v2f a = *reinterpret_cast<const v2f*>(&la[row * LDSPAD + k0 + 2 * half]);

        // B fragments: B[k][n] = W_gen[k0+2*half (+1)][n0 + t*16 + row]
        const float* wp = Wg + (size_t)(k0 + 2 * half) * OUTDIM + n0 + row;
        // Keep the W_gen stream ahead: next k-step's rows (gfx1250 global_prefetch_b8).
        __builtin_prefetch(wp + (size_t)4 * OUTDIM, 0, 1);
        v2f b0 = { wp[ 0], wp[ 0 + OUTDIM] };
        v2f b1 = { wp[16], wp[16 + OUTDIM] };
        v2f b2 = { wp[32], wp[32 + OUTDIM] };
        v2f b3 = { wp[48], wp[48 + OUTDIM] };

        acc0 = wmma_f32_4(a, b0, acc0);
        acc1 = wmma_f32_4(a, b1, acc1);
        acc2 = wmma_f32_4(a, b2, acc2);
        acc3 = wmma_f32_4(a, b3, acc3);
    }

    // Bias + store (chunk-relative rows). D layout: lane -> n = row,
    // VGPR vr -> m = vr + 8*half.
    #pragma unroll
    for (int t = 0; t < 4; ++t) {
        const v8f acc = (t == 0) ? acc0 : (t == 1) ? acc1 : (t == 2) ? acc2 : acc3;
        const int col = n0 + t * 16 + row;
        const float bias = bg[col];
        #pragma unroll
        for (int vr = 0; vr < 8; ++vr) {
            const int mrel = mrel0 + vr + 8 * half;
            w_out[(size_t)mrel * OUTDIM + col] = acc[vr] + bias;
        }
    }
}

// ---------------------------------------------------------------------------
// Stage 2: per-query mixing + LayerNorm + ReLU.
// grid = chunk_rows blocks, block = 256 (8 waves). Wave w: g = w&3, ptile = w>>2.
// LDS buffer is reused: phase A holds x tile, phase B holds mixed tile.
// ---------------------------------------------------------------------------
__global__ __launch_bounds__(256) void acm_mix_ln_kernel(
    const float* __restrict__ x,      // [2400, 32, 256]
    const float* __restrict__ w_ws,   // [chunk_rows, 16384]
    const float* __restrict__ gamma,  // [256]
    const float* __restrict__ beta,   // [256]
    float* __restrict__ out,          // [2400, 32, 256]
    int bn0)
{
    __shared__ float buf[PNUM * LDSPAD];   // 33.3 KB, reused x -> mixed
    __shared__ float smean[PNUM];
    __shared__ float srstd[PNUM];

    const int tid  = threadIdx.x;
    const int lane = tid & 31;
    const int wv   = tid >> 5;        // 0..7
    const int g    = wv & 3;          // channel group
    const int pt   = wv >> 2;         // p-tile (0/1)
    const int bn   = bn0 + blockIdx.x;

    const size_t xbase = (size_t)bn * PNUM * HIDDEN;

    // Phase A: stage x tile [32 x 256] into LDS.
#if HAVE_ASYNC_LDS
    // 32 KB = 2048 x 16B chunks, 8 per thread, consecutive tid -> coalesced.
    for (int ci = tid; ci < 2048; ci += 256) {
        const int f = ci * 4;
        const int r = f >> 8, c = f & 255;
        __builtin_amdgcn_global_load_async_to_lds_b128(
            to_global_v4(x + xbase + f),
            to_lds_v4(&buf[r * LDSPAD + c]), 0, 0);
    }
    async_wait_all();
#else
    #pragma unroll 4
    for (int i = tid; i < PNUM * HIDDEN; i += 256) {
        const int r = i >> 8, c = i & 255;
        buf[r * LDSPAD + c] = x[xbase + i];
    }
#endif
    __syncthreads();

    const int row  = lane & 15;       // A: p within tile / B,D: e within tile
    const int half = lane >> 4;
    const int p    = pt * 16 + row;

    const float* wq = w_ws + (size_t)blockIdx.x * OUTDIM + g * 4096;  // w[g][e][d]

    v8f acc0{}, acc1{}, acc2{}, acc3{};

    #pragma unroll 4
    for (int k0 = 0; k0 < 64; k0 += 4) {
        const int k = k0 + 2 * half;
        // A: x[p][g*64 + k], +1 (8B LDS load)
        const v2f a = *reinterpret_cast<const v2f*>(&buf[p * LDSPAD + g * 64 + k]);
        // B[k][e] = w[e][k]: rows of w contiguous in d -> 8B global loads
        const float* wp = wq + (size_t)row * 64 + k;
        const v2f b0 = *reinterpret_cast<const v2f*>(wp);            // e tile 0
        const v2f b1 = *reinterpret_cast<const v2f*>(wp + 16 * 64);  // e tile 1
        const v2f b2 = *reinterpret_cast<const v2f*>(wp + 32 * 64);  // e tile 2
        const v2f b3 = *reinterpret_cast<const v2f*>(wp + 48 * 64);  // e tile 3
        acc0 = wmma_f32_4(a, b0, acc0);
        acc1 = wmma_f32_4(a, b1, acc1);
        acc2 = wmma_f32_4(a, b2, acc2);
        acc3 = wmma_f32_4(a, b3, acc3);
    }
    __syncthreads();   // all x reads complete before buffer reuse

    // Phase B: scatter mixed tiles into LDS. D layout: lane -> col = row,
    // VGPR vr -> row-in-tile = vr + 8*half.
    #pragma unroll
    for (int t = 0; t < 4; ++t) {
        const v8f acc = (t == 0) ? acc0 : (t == 1) ? acc1 : (t == 2) ? acc2 : acc3;
        const int c = g * 64 + t * 16 + row;
        #pragma unroll
        for (int vr = 0; vr < 8; ++vr) {
            const int pr = pt * 16 + vr + 8 * half;
            buf[pr * LDSPAD + c] = acc[vr];
        }
    }
    __syncthreads();

    // LayerNorm stats: 8 lanes per row, 32 channels each, shfl_xor reduce.
    {
        const int r = tid >> 3, sub = tid & 7;
        float s = 0.f, ss = 0.f;
        #pragma unroll 8
        for (int j = 0; j < 32; ++j) {
            const float v = buf[r * LDSPAD + sub * 32 + j];
            s += v; ss += v * v;
        }
        s  += __shfl_xor(s, 1, 32);  ss += __shfl_xor(ss, 1, 32);
        s  += __shfl_xor(s, 2, 32);  ss += __shfl_xor(ss, 2, 32);
        s  += __shfl_xor(s, 4, 32);  ss += __shfl_xor(ss, 4, 32);
        if (sub == 0) {
            const float mean = s * (1.0f / 256.0f);
            const float var  = ss * (1.0f / 256.0f) - mean * mean;
            smean[r] = mean;
            srstd[r] = rsqrtf(var + 1e-5f);
        }
    }
    __syncthreads();

    // Normalize + ReLU + coalesced store.
    #pragma unroll 4
    for (int i = tid; i < PNUM * HIDDEN; i += 256) {
        const int pr = i >> 8, c = i & 255;
        const float v = (buf[pr * LDSPAD + c] - smean[pr]) * srstd[pr] * gamma[c] + beta[c];
        out[xbase + i] = fmaxf(v, 0.0f);
    }
}

// ---------------------------------------------------------------------------
// Launcher: chunk the 157 MB `w` intermediate through the workspace; stage1
// and stage2 of each chunk are stream-ordered so the intermediate stays hot
// in the 192 MB L2.
// ---------------------------------------------------------------------------
extern "C" void kernel_launch(void* const* d_in, const int* in_sizes, int n_in,
                              void* d_out, int out_size, void* d_ws, size_t ws_size,
                              hipStream_t stream) {
    const float* x     = (const float*)d_in[0];
    const float* qc    = (const float*)d_in[1];
    const float* Wgen  = (const float*)d_in[2];
    const float* bgen  = (const float*)d_in[3];
    const float* gamma = (const float*)d_in[4];
    const float* beta  = (const float*)d_in[5];
    float* out = (float*)d_out;
    float* ws  = (float*)d_ws;

    const size_t bytesPerRow = (size_t)OUTDIM * sizeof(float);
    int chunk = (int)(ws_size / bytesPerRow);
    chunk &= ~15;                 // multiple of 16 rows
    if (chunk < 16)   chunk = 16;
    if (chunk > MTOT) chunk = MTOT;

    for (int m0 = 0; m0 < MTOT; m0 += chunk) {
        const int rows = (MTOT - m0 < chunk) ? (MTOT - m0) : chunk;  // multiple of 16
        dim3 g1(OUTDIM / 64, rows / 16);
        acm_wgen_kernel<<<g1, 32, 0, stream>>>(qc, Wgen, bgen, ws, m0);
        acm_mix_ln_kernel<<<dim3(rows), 256, 0, stream>>>(x, ws, gamma, beta, out, m0);
    }
}